// GinNet_64991445123450
// MI455X (gfx1250) — compile-verified
//
#include <hip/hip_runtime.h>

typedef float v2f __attribute__((ext_vector_type(2)));
typedef float v8f __attribute__((ext_vector_type(8)));

#define NN 100000
#define NE 3200000
#define NG 1000
#define HD 64

static_assert(NN % 16 == 0, "wave tiles must cover rows exactly");

__global__ void k_zero(float* p, long n) {
  long i = (long)blockIdx.x * blockDim.x + threadIdx.x;
  long stride = (long)gridDim.x * blockDim.x;
  for (; i < n; i += stride) p[i] = 0.0f;
}

// layer-1 aggregation: F_IN = 3
__global__ void k_scatter3(const float* __restrict__ x,
                           const long long* __restrict__ src,
                           const long long* __restrict__ dst,
                           float* __restrict__ agg) {
  long t = (long)blockIdx.x * blockDim.x + threadIdx.x;
  long e = t >> 2; int f = (int)(t & 3);
  if (e >= NE || f >= 3) return;
  int s = (int)src[e]; int d = (int)dst[e];
  atomicAdd(&agg[(long)d * 3 + f], x[(long)s * 3 + f]);
}

// hidden-layer aggregation: 64 features, lane-per-feature (coalesced 128B/wave)
__global__ void k_scatter64(const float* __restrict__ h,
                            const long long* __restrict__ src,
                            const long long* __restrict__ dst,
                            float* __restrict__ agg) {
  long t = (long)blockIdx.x * blockDim.x + threadIdx.x;
  long e = t >> 6; int f = (int)(t & 63);
  if (e >= NE) return;
  int s = (int)src[e]; int d = (int)dst[e];
  atomicAdd(&agg[(long)d * HD + f], h[(long)s * HD + f]);
}

// layer 1: out = relu(((1+eps)*x + agg) @ W1 + b1), K = 3 padded to 4
__global__ void k_gin_gemm_fin(const float* __restrict__ x,
                               const float* __restrict__ agg,
                               const float* __restrict__ eps,
                               const float* __restrict__ W,   // 3 x 64
                               const float* __restrict__ bias,
                               float* __restrict__ out) {
  __shared__ float sW[4 * HD];
  __shared__ float sB[HD];
  int tid = threadIdx.x;
  if (tid < 4 * HD) sW[tid] = (tid < 3 * HD) ? W[tid] : 0.0f;
  if (tid < HD) sB[tid] = bias[tid];
  __syncthreads();
  int wave = tid >> 5, lane = tid & 31;
  int half = lane >> 4, mr = lane & 15;
  int m0 = (blockIdx.x * 8 + wave) * 16;
  if (m0 >= NN) return;          // wave-uniform: EXEC stays all-ones
  float ep = 1.0f + *eps;
  int row = m0 + mr;
  v2f a;
  if (half == 0) {
    a.x = ep * x[(long)row * 3 + 0] + agg[(long)row * 3 + 0];
    a.y = ep * x[(long)row * 3 + 1] + agg[(long)row * 3 + 1];
  } else {
    a.x = ep * x[(long)row * 3 + 2] + agg[(long)row * 3 + 2];
    a.y = 0.0f;
  }
#pragma unroll
  for (int nt = 0; nt < 4; ++nt) {
    int col = nt * 16 + mr;
    v2f b;
    b.x = sW[(2 * half + 0) * HD + col];
    b.y = sW[(2 * half + 1) * HD + col];
    v8f c = {};
    c = __builtin_amdgcn_wmma_f32_16x16x4_f32(false, a, false, b, (short)0, c,
                                              false, false);
    float bb = sB[col];
#pragma unroll
    for (int v = 0; v < 8; ++v) {
      int m = m0 + v + 8 * half;  // C/D layout: VGPR v -> row v + 8*half
      float val = c[v] + bb;
      out[(long)m * HD + col] = val > 0.0f ? val : 0.0f;
    }
  }
}

// hidden layers: out = relu(((1+eps)*hin + agg) @ W + b), 64x64
__global__ void k_gin_gemm64(const float* __restrict__ hin,
                             const float* __restrict__ agg,
                             const float* __restrict__ eps,
                             const float* __restrict__ W,   // 64 x 64
                             const float* __restrict__ bias,
                             float* __restrict__ out) {
  __shared__ float sW[HD * HD];
  __shared__ float sB[HD];
  int tid = threadIdx.x;
  for (int i = tid; i < HD * HD; i += blockDim.x) sW[i] = W[i];
  if (tid < HD) sB[tid] = bias[tid];
  __syncthreads();
  int wave = tid >> 5, lane = tid & 31;
  int half = lane >> 4, mr = lane & 15;
  int m0 = (blockIdx.x * 8 + wave) * 16;
  if (m0 >= NN) return;          // wave-uniform: EXEC stays all-ones
  float ep = 1.0f + *eps;
  int row = m0 + mr;
  const float* xr = hin + (long)row * HD;
  const float* ar = agg + (long)row * HD;
  v8f acc[4] = {v8f{}, v8f{}, v8f{}, v8f{}};
#pragma unroll
  for (int k = 0; k < HD; k += 4) {
    int kc = k + 2 * half;  // A layout: lanes 0-15 hold K,K+1; 16-31 hold K+2,K+3
    v2f a;
    a.x = ep * xr[kc + 0] + ar[kc + 0];
    a.y = ep * xr[kc + 1] + ar[kc + 1];
#pragma unroll
    for (int nt = 0; nt < 4; ++nt) {
      int col = nt * 16 + mr;
      v2f b;
      b.x = sW[(kc + 0) * HD + col];
      b.y = sW[(kc + 1) * HD + col];
      acc[nt] = __builtin_amdgcn_wmma_f32_16x16x4_f32(false, a, false, b,
                                                      (short)0, acc[nt],
                                                      false, false);
    }
  }
#pragma unroll
  for (int nt = 0; nt < 4; ++nt) {
    int col = nt * 16 + mr;
    float bb = sB[col];
#pragma unroll
    for (int v = 0; v < 8; ++v) {
      int m = m0 + v + 8 * half;
      float val = acc[nt][v] + bb;
      out[(long)m * HD + col] = val > 0.0f ? val : 0.0f;
    }
  }
}

__global__ void k_pool(const float* __restrict__ h,
                       const long long* __restrict__ batch,
                       float* __restrict__ psum, float* __restrict__ pcnt) {
  long t = (long)blockIdx.x * blockDim.x + threadIdx.x;
  long n = t >> 6; int f = (int)(t & 63);
  if (n >= NN) return;
  int g = (int)batch[n];
  atomicAdd(&psum[(long)g * HD + f], h[n * HD + f]);
  if (f == 0) atomicAdd(&pcnt[g], 1.0f);
}

__global__ void k_head(const float* __restrict__ psum,
                       const float* __restrict__ pcnt,
                       const float* __restrict__ Wf1, const float* __restrict__ bf1,
                       const float* __restrict__ Wf2, const float* __restrict__ bf2,
                       float* __restrict__ out) {
  int g = blockIdx.x * blockDim.x + threadIdx.x;
  if (g >= NG) return;
  float cnt = pcnt[g];
  if (cnt < 1.0f) cnt = 1.0f;
  float inv = 1.0f / cnt;
  float pooled[HD];
  for (int k = 0; k < HD; ++k) pooled[k] = psum[(long)g * HD + k] * inv;
  float o = bf2[0];
  for (int j = 0; j < 10; ++j) {
    float s = bf1[j];
    for (int k = 0; k < HD; ++k) s += pooled[k] * Wf1[k * 10 + j];
    if (s < 0.0f) s = 0.0f;
    o += s * Wf2[j];
  }
  out[g] = o;
}

extern "C" void kernel_launch(void* const* d_in, const int* in_sizes, int n_in,
                              void* d_out, int out_size, void* d_ws, size_t ws_size,
                              hipStream_t stream) {
  (void)in_sizes; (void)n_in; (void)out_size; (void)ws_size;
  const float*     x    = (const float*)d_in[0];
  const long long* eidx = (const long long*)d_in[1];
  const long long* bat  = (const long long*)d_in[2];
  const float* eps1 = (const float*)d_in[3];
  const float* eps2 = (const float*)d_in[4];
  const float* eps3 = (const float*)d_in[5];
  const float* W1 = (const float*)d_in[6];
  const float* b1 = (const float*)d_in[7];
  const float* W2 = (const float*)d_in[8];
  const float* b2 = (const float*)d_in[9];
  const float* W3 = (const float*)d_in[10];
  const float* b3 = (const float*)d_in[11];
  const float* Wf1 = (const float*)d_in[12];
  const float* bf1 = (const float*)d_in[13];
  const float* Wf2 = (const float*)d_in[14];
  const float* bf2 = (const float*)d_in[15];

  const long long* src = eidx;        // edge_index[0]
  const long long* dst = eidx + NE;   // edge_index[1]

  char* ws = (char*)d_ws;
  size_t szH = (size_t)NN * HD * sizeof(float);   // 25.6 MB
  float* agg  = (float*)(ws);
  float* hA   = (float*)(ws + szH);
  float* hB   = (float*)(ws + 2 * szH);
  float* psum = (float*)(ws + 3 * szH);
  float* pcnt = psum + (size_t)NG * HD;

  const int gemmBlocks = (NN + 127) / 128;   // 8 waves x 16 rows per block

  // ---- layer 1 ----
  k_zero<<<2048, 256, 0, stream>>>(agg, (long)NN * 3);
  k_scatter3<<<(NE * 4) / 256, 256, 0, stream>>>(x, src, dst, agg);
  k_gin_gemm_fin<<<gemmBlocks, 256, 0, stream>>>(x, agg, eps1, W1, b1, hA);

  // ---- layer 2 ----
  k_zero<<<4096, 256, 0, stream>>>(agg, (long)NN * HD);
  k_scatter64<<<(int)(((long)NE * HD) / 256), 256, 0, stream>>>(hA, src, dst, agg);
  k_gin_gemm64<<<gemmBlocks, 256, 0, stream>>>(hA, agg, eps2, W2, b2, hB);

  // ---- layer 3 (reuse hA as output) ----
  k_zero<<<4096, 256, 0, stream>>>(agg, (long)NN * HD);
  k_scatter64<<<(int)(((long)NE * HD) / 256), 256, 0, stream>>>(hB, src, dst, agg);
  k_gin_gemm64<<<gemmBlocks, 256, 0, stream>>>(hB, agg, eps3, W3, b3, hA);

  // ---- pool + head ----
  k_zero<<<64, 256, 0, stream>>>(psum, (long)NG * HD + NG);
  k_pool<<<(NN * 64) / 256, 256, 0, stream>>>(hA, bat, psum, pcnt);
  k_head<<<(NG + 255) / 256, 256, 0, stream>>>(psum, pcnt, Wf1, bf1, Wf2, bf2,
                                               (float*)d_out);
}